// M4_5514738008543
// MI455X (gfx1250) — compile-verified
//
#include <hip/hip_runtime.h>

// ---------------------------------------------------------------------------
// GINEConv x3 + BN + LeakyReLU + global-add-pool + final linear, fp32 WMMA
// (V_WMMA_F32_16X16X4_F32) on gfx1250 (wave32).
// ---------------------------------------------------------------------------

typedef __attribute__((ext_vector_type(2))) float v2f;
typedef __attribute__((ext_vector_type(8))) float v8f;

#define N_NODES 50000
#define N_EDGES 800000
#define N_GRAPH 2500
#define DIM     96      // node feature / conv width
#define EDIM    16      // edge feature
#define NT      6       // DIM / 16 column tiles per wave
#define NEG_SLOPE 0.01f
#define BN_EPS  1e-5f

__device__ __forceinline__ v8f wmma_f32(v2f a, v2f b, v8f c) {
  // D = A(16x4,f32) * B(4x16,f32) + C(16x16,f32)
  return __builtin_amdgcn_wmma_f32_16x16x4_f32(
      /*neg_a=*/false, a, /*neg_b=*/false, b,
      /*c_mod=*/(short)0, c, /*reuse_a=*/false, /*reuse_b=*/false);
}

__device__ __forceinline__ float leaky(float v) {
  return v > 0.0f ? v : NEG_SLOPE * v;
}

// ---------------------------------------------------------------------------
__global__ __launch_bounds__(256) void zero_kernel(float* __restrict__ p, int n) {
  int i = blockIdx.x * blockDim.x + threadIdx.x;
  if (i < n) p[i] = 0.0f;
}

// ---------------------------------------------------------------------------
// Edge kernel: e = edge_attr @ We + be (WMMA, K=16), msg = relu(x[src]+e),
// atomic scatter-add into aggr[dst]. One wave = 16 edges x 96 cols.
// ---------------------------------------------------------------------------
__global__ __launch_bounds__(256) void edge_conv_kernel(
    const float* __restrict__ x,      // [N, 96]
    const int*   __restrict__ ei,     // [2, E] (src row 0, dst row 1)
    const float* __restrict__ ea,     // [E, 16]
    const float* __restrict__ We,     // [16, 96] layer slice
    const float* __restrict__ be,     // [96]
    float*       __restrict__ aggr)   // [N, 96]
{
  // LDS: We in (k,k+1)-paired layout so B fragments are one ds_load_b64.
  __shared__ float sW[(EDIM / 2) * DIM * 2];   // 1536 floats
  __shared__ float sb[DIM];
  for (int i = threadIdx.x; i < EDIM * DIM; i += blockDim.x) {
    int k = i / DIM, col = i % DIM;
    sW[(k >> 1) * (2 * DIM) + col * 2 + (k & 1)] = We[i];
  }
  for (int i = threadIdx.x; i < DIM; i += blockDim.x) sb[i] = be[i];
  __syncthreads();

  const int wave = threadIdx.x >> 5;
  const int lane = threadIdx.x & 31;
  const int hl   = lane >> 4;     // half-wave selector
  const int ln   = lane & 15;     // M (A) / N (B,C) within tile
  const int tile = blockIdx.x * (blockDim.x >> 5) + wave;
  const int e0   = tile * 16;
  if (e0 >= N_EDGES) return;

  // A fragments: 16x16 tile of edge_attr; lane ln holds row ln, cols k0+2*hl,+1
  const float* arow = ea + (e0 + ln) * EDIM;
  v2f afrag[4];
#pragma unroll
  for (int ks = 0; ks < 4; ++ks)
    afrag[ks] = *(const v2f*)(arow + ks * 4 + 2 * hl);

  // src/dst for the 8 C rows this lane owns (rows r + 8*hl)
  int srcs[8], dsts[8];
#pragma unroll
  for (int r = 0; r < 8; ++r) {
    int e   = e0 + r + 8 * hl;
    srcs[r] = ei[e] * DIM;
    dsts[r] = ei[N_EDGES + e] * DIM;
  }

#pragma unroll
  for (int t = 0; t < NT; ++t) {
    const int col = t * 16 + ln;
    const float bias = sb[col];
    v8f c = {bias, bias, bias, bias, bias, bias, bias, bias};
#pragma unroll
    for (int ks = 0; ks < 4; ++ks) {
      int kk = ks * 4 + 2 * hl;
      v2f b = *(const v2f*)(&sW[(kk >> 1) * (2 * DIM) + col * 2]);
      c = wmma_f32(afrag[ks], b, c);
    }
#pragma unroll
    for (int r = 0; r < 8; ++r) {
      float msg = x[srcs[r] + col] + c[r];
      msg = msg > 0.0f ? msg : 0.0f;               // relu
      atomicAdd(&aggr[dsts[r] + col], msg);        // global_atomic_add_f32
    }
  }
}

// ---------------------------------------------------------------------------
// Node GEMM, 16 rows x 96 cols per wave, K=96 via WMMA f32 16x16x4.
// MODE 0: A = (1+eps)*x + aggr          (GINE residual fused)
// MODE 1: A = leaky(t1*scale + shift)   (BN1 + LeakyReLU fused)
// Writes out = A@W + bias; optionally accumulates per-column sum/sumsq.
// ---------------------------------------------------------------------------
template <int MODE>
__global__ __launch_bounds__(256) void gemm_kernel(
    const float* __restrict__ A0,     // x (MODE0) or t1 (MODE1)
    const float* __restrict__ A1,     // aggr (MODE0)
    const float* __restrict__ epsp,   // &eps[l] (MODE0)
    const float* __restrict__ scale,  // [96] (MODE1)
    const float* __restrict__ shift,  // [96] (MODE1)
    const float* __restrict__ W,      // [96, 96]
    const float* __restrict__ bias,   // [96]
    float*       __restrict__ out,    // [N, 96]
    float*       __restrict__ gsum,   // [96]
    float*       __restrict__ gsq,    // [96]
    int do_stats)
{
  __shared__ float sW[(DIM / 2) * DIM * 2];   // 9216 floats = 36 KB, paired
  __shared__ float sb[DIM];
  __shared__ float ssc[DIM];
  __shared__ float ssh[DIM];
  __shared__ float ssum[DIM];
  __shared__ float ssq[DIM];
  for (int i = threadIdx.x; i < DIM * DIM; i += blockDim.x) {
    int k = i / DIM, col = i % DIM;
    sW[(k >> 1) * (2 * DIM) + col * 2 + (k & 1)] = W[i];
  }
  if (threadIdx.x < DIM) {
    sb[threadIdx.x] = bias[threadIdx.x];
    if (MODE == 1) {
      ssc[threadIdx.x] = scale[threadIdx.x];
      ssh[threadIdx.x] = shift[threadIdx.x];
    }
    ssum[threadIdx.x] = 0.0f;
    ssq[threadIdx.x]  = 0.0f;
  }
  __syncthreads();

  const float epsv = (MODE == 0) ? (1.0f + epsp[0]) : 0.0f;
  const int wave = threadIdx.x >> 5;
  const int lane = threadIdx.x & 31;
  const int hl   = lane >> 4;
  const int ln   = lane & 15;
  const int rowbase = (blockIdx.x * (blockDim.x >> 5) + wave) * 16;
  const bool active = rowbase < N_NODES;

  if (active) {
    const int  m      = rowbase + ln;
    const bool mvalid = (m < N_NODES);
    const int  mrow   = mvalid ? m : 0;           // clamp, zero via mscale
    const float mscale = mvalid ? 1.0f : 0.0f;
    const float* a0row = A0 + mrow * DIM;
    const float* a1row = (MODE == 0) ? (A1 + mrow * DIM) : A0;

    v8f c[NT];
#pragma unroll
    for (int t = 0; t < NT; ++t) {
      float bv = sb[t * 16 + ln];
      c[t] = (v8f){bv, bv, bv, bv, bv, bv, bv, bv};
    }

    for (int k0 = 0; k0 < DIM; k0 += 4) {
      const int kk = k0 + 2 * hl;
      v2f a;
      if (MODE == 0) {
        v2f xv = *(const v2f*)(a0row + kk);
        v2f gv = *(const v2f*)(a1row + kk);
        a.x = (epsv * xv.x + gv.x) * mscale;
        a.y = (epsv * xv.y + gv.y) * mscale;
      } else {
        v2f tv = *(const v2f*)(a0row + kk);
        a.x = leaky(tv.x * ssc[kk] + ssh[kk]) * mscale;
        a.y = leaky(tv.y * ssc[kk + 1] + ssh[kk + 1]) * mscale;
      }
#pragma unroll
      for (int t = 0; t < NT; ++t) {
        v2f b = *(const v2f*)(&sW[(kk >> 1) * (2 * DIM) + (t * 16 + ln) * 2]);
        c[t] = wmma_f32(a, b, c[t]);
      }
    }

    // Store + per-column BN statistics (sum, sumsq over rows).
#pragma unroll
    for (int t = 0; t < NT; ++t) {
      const int col = t * 16 + ln;
      float s = 0.0f, q = 0.0f;
#pragma unroll
      for (int r = 0; r < 8; ++r) {
        int row = rowbase + r + 8 * hl;
        if (row < N_NODES) {
          float v = c[t][r];
          out[row * DIM + col] = v;
          s += v;
          q += v * v;
        }
      }
      s += __shfl_xor(s, 16, 32);   // fold halves: lanes n and n+16 share col n
      q += __shfl_xor(q, 16, 32);
      if (do_stats && hl == 0) {
        atomicAdd(&ssum[col], s);   // ds_add_f32
        atomicAdd(&ssq[col], q);
      }
    }
  }
  __syncthreads();
  if (do_stats && threadIdx.x < DIM) {
    atomicAdd(&gsum[threadIdx.x], ssum[threadIdx.x]);
    atomicAdd(&gsq[threadIdx.x], ssq[threadIdx.x]);
  }
}

// ---------------------------------------------------------------------------
__global__ void bn_finalize_kernel(const float* __restrict__ gsum,
                                   const float* __restrict__ gsq,
                                   const float* __restrict__ g,
                                   const float* __restrict__ bt,
                                   float* __restrict__ scale,
                                   float* __restrict__ shift, float invn) {
  int c = threadIdx.x;
  if (c < DIM) {
    float mean = gsum[c] * invn;
    float var  = gsq[c] * invn - mean * mean;   // biased variance
    float sc   = g[c] * rsqrtf(var + BN_EPS);
    scale[c] = sc;
    shift[c] = bt[c] - mean * sc;
  }
}

__global__ __launch_bounds__(256) void bn_apply_kernel(
    float* __restrict__ x, const float* __restrict__ scale,
    const float* __restrict__ shift) {
  int i = blockIdx.x * blockDim.x + threadIdx.x;
  if (i < N_NODES * DIM) {
    int col = i % DIM;
    x[i] = leaky(x[i] * scale[col] + shift[col]);
  }
}

__global__ __launch_bounds__(256) void pool_kernel(
    const float* __restrict__ x, const int* __restrict__ batch,
    float* __restrict__ pool) {
  int i = blockIdx.x * blockDim.x + threadIdx.x;
  if (i < N_NODES * DIM) {
    int row = i / DIM, col = i % DIM;
    atomicAdd(&pool[batch[row] * DIM + col], x[i]);
  }
}

__global__ __launch_bounds__(256) void final_kernel(
    const float* __restrict__ x, const float* __restrict__ pool,
    const int* __restrict__ batch, const float* __restrict__ fW,
    const float* __restrict__ fb, float* __restrict__ out) {
  int i = blockIdx.x * blockDim.x + threadIdx.x;
  if (i < N_NODES) {
    const float* xr = x + i * DIM;
    const float* pr = pool + batch[i] * DIM;
    float acc = fb[0];
#pragma unroll 8
    for (int k = 0; k < DIM; ++k) acc += xr[k] * fW[k];
#pragma unroll 8
    for (int k = 0; k < DIM; ++k) acc += pr[k] * fW[DIM + k];
    out[i] = acc;
  }
}

// ---------------------------------------------------------------------------
static inline int cdiv(int a, int b) { return (a + b - 1) / b; }

extern "C" void kernel_launch(void* const* d_in, const int* in_sizes, int n_in,
                              void* d_out, int out_size, void* d_ws,
                              size_t ws_size, hipStream_t stream) {
  (void)in_sizes; (void)n_in; (void)out_size; (void)ws_size;
  const float* x_in  = (const float*)d_in[0];
  const int*   ei    = (const int*)d_in[1];     // edge_index [2,E]
  const float* ea    = (const float*)d_in[2];   // edge_attr [E,16]
  const int*   batch = (const int*)d_in[3];
  const float* We    = (const float*)d_in[4];   // [L,16,96]
  const float* be    = (const float*)d_in[5];   // [L,96]
  const float* eps   = (const float*)d_in[6];   // [L]
  const float* W1    = (const float*)d_in[7];   // [L,96,96]
  const float* b1    = (const float*)d_in[8];
  const float* g1    = (const float*)d_in[9];
  const float* bt1   = (const float*)d_in[10];
  const float* W2    = (const float*)d_in[11];
  const float* b2    = (const float*)d_in[12];
  const float* g_out = (const float*)d_in[13];
  const float* bt_out= (const float*)d_in[14];
  const float* fW    = (const float*)d_in[15];  // [192]
  const float* fb    = (const float*)d_in[16];  // [1]
  float* out = (float*)d_out;

  // Workspace layout (floats): aggr[N*96] | t1[N*96] | xa[N*96] | pool[G*96] | stats[8*96]
  float* ws   = (float*)d_ws;
  float* aggr = ws;
  float* t1   = aggr + (size_t)N_NODES * DIM;
  float* xa   = t1 + (size_t)N_NODES * DIM;
  float* pool = xa + (size_t)N_NODES * DIM;
  float* st   = pool + (size_t)N_GRAPH * DIM;
  float* sum1 = st;            float* sq1 = st + DIM;
  float* sc1  = st + 2 * DIM;  float* sh1 = st + 3 * DIM;
  float* sumo = st + 4 * DIM;  float* sqo = st + 5 * DIM;
  float* sco  = st + 6 * DIM;  float* sho = st + 7 * DIM;

  const int ND = N_NODES * DIM;                 // 4.8M
  const int edge_blocks = cdiv(N_EDGES / 16, 8);   // 6250 (8 waves/block)
  const int gemm_blocks = cdiv(cdiv(N_NODES, 16), 8);  // 391
  const float invn = 1.0f / (float)N_NODES;

  const float* xcur = x_in;
  for (int l = 0; l < 3; ++l) {
    zero_kernel<<<cdiv(ND, 256), 256, 0, stream>>>(aggr, ND);
    zero_kernel<<<cdiv(8 * DIM, 256), 256, 0, stream>>>(st, 8 * DIM);

    edge_conv_kernel<<<edge_blocks, 256, 0, stream>>>(
        xcur, ei, ea, We + l * EDIM * DIM, be + l * DIM, aggr);

    gemm_kernel<0><<<gemm_blocks, 256, 0, stream>>>(
        xcur, aggr, eps + l, nullptr, nullptr, W1 + l * DIM * DIM,
        b1 + l * DIM, t1, sum1, sq1, /*do_stats=*/1);

    bn_finalize_kernel<<<1, 128, 0, stream>>>(sum1, sq1, g1 + l * DIM,
                                              bt1 + l * DIM, sc1, sh1, invn);

    const int do_outer = (l != 2);
    // GEMM2 never reads xcur, so writing xa in place (layers 1,2) is safe.
    gemm_kernel<1><<<gemm_blocks, 256, 0, stream>>>(
        t1, nullptr, nullptr, sc1, sh1, W2 + l * DIM * DIM, b2 + l * DIM, xa,
        sumo, sqo, do_outer);

    if (do_outer) {
      bn_finalize_kernel<<<1, 128, 0, stream>>>(
          sumo, sqo, g_out + l * DIM, bt_out + l * DIM, sco, sho, invn);
      bn_apply_kernel<<<cdiv(ND, 256), 256, 0, stream>>>(xa, sco, sho);
    }
    xcur = xa;
  }

  zero_kernel<<<cdiv(N_GRAPH * DIM, 256), 256, 0, stream>>>(pool, N_GRAPH * DIM);
  pool_kernel<<<cdiv(ND, 256), 256, 0, stream>>>(xcur, batch, pool);
  final_kernel<<<cdiv(N_NODES, 256), 256, 0, stream>>>(xcur, pool, batch, fW,
                                                       fb, out);
}